// solvgnn_binary_40785009443204
// MI455X (gfx1250) — compile-verified
//
#include <hip/hip_runtime.h>
#include <hip/hip_bf16.h>

// ---------------- problem constants (match reference) ----------------
#define BATCH 2048
#define NPM   30
#define NNODE (BATCH * NPM)   // 61440 nodes per side
#define EPM   120
#define INDIM 74
#define HDIM  128
#define NC    2

typedef __attribute__((ext_vector_type(2))) float v2f;
typedef __attribute__((ext_vector_type(8))) float v8f;

__device__ __forceinline__ int imax(int a, int b) { return a > b ? a : b; }
__device__ __forceinline__ int imin(int a, int b) { return a < b ? a : b; }

#define WMMA_F32(A_, B_, C_) \
    __builtin_amdgcn_wmma_f32_16x16x4_f32(false, (A_), false, (B_), (short)0, (C_), false, false)

// =====================================================================
// fp32 WMMA GEMM:  C[M x Nc] = epilogue(A[M x K] @ B[K x Nc] + bias)
// One wave computes a 16x64 tile (4x V_WMMA_F32_16X16X4_F32 per K-step;
// A fragment reused across the 4 N-subtiles).
// Distance-2 software pipeline: fragments for steps i and i+1 stay
// resident; step i+2's loads are issued before step i's WMMAs, so the
// WMMAs wait only on loads issued two steps earlier (partial loadcnt
// waits -> load latency overlapped with matrix ops).
// Requirements: M % 16 == 0, Nc % 64 == 0, K even >= 4. (All call sites OK.)
// =====================================================================
__global__ void gemm_wmma_f32(const float* __restrict__ A,
                              const float* __restrict__ Bm,
                              const float* __restrict__ bias,
                              float* __restrict__ C,
                              int M, int Nc, int K, int doRelu)
{
    const int lane   = threadIdx.x & 31;
    const int waveId = (blockIdx.x * (blockDim.x >> 5)) + (threadIdx.x >> 5);
    const int tilesN = Nc >> 6;                   // 64-wide wave tiles
    const int total  = (M >> 4) * tilesN;
    if (waveId >= total) return;                  // wave-uniform: EXEC stays all-ones

    const int tm   = waveId / tilesN;
    const int tn   = waveId - tm * tilesN;
    const int row0 = tm << 4;
    const int col0 = tn << 6;
    const int half = lane >> 4;                   // 0 -> k0,k0+1 ; 1 -> k0+2,k0+3
    const int l    = lane & 15;
    const int arow = row0 + l;                    // A row held by this lane
    const int kofs = half << 1;

    v8f acc0 = {0.f,0.f,0.f,0.f,0.f,0.f,0.f,0.f};
    v8f acc1 = acc0, acc2 = acc0, acc3 = acc0;

    const float* ap = A  + (size_t)arow * K + kofs;          // 8B aligned (K even)
    const float* bp = Bm + (size_t)kofs * Nc + col0 + l;
    const size_t bstep = (size_t)Nc << 2;

    const int K4    = K & ~3;
    const int steps = K4 >> 2;                    // >= 1 for all call sites

    // ---- prologue: load fragments for steps 0 and 1 ----
    v2f a0 = *reinterpret_cast<const v2f*>(ap);  ap += 4;
    v2f b00, b01, b02, b03;
    b00.x = bp[0];  b00.y = bp[Nc];
    b01.x = bp[16]; b01.y = bp[Nc + 16];
    b02.x = bp[32]; b02.y = bp[Nc + 32];
    b03.x = bp[48]; b03.y = bp[Nc + 48];
    bp += bstep;

    v2f a1 = a0;
    v2f b10 = b00, b11 = b01, b12 = b02, b13 = b03;
    if (steps > 1) {                              // uniform branch
        a1 = *reinterpret_cast<const v2f*>(ap);  ap += 4;
        b10.x = bp[0];  b10.y = bp[Nc];
        b11.x = bp[16]; b11.y = bp[Nc + 16];
        b12.x = bp[32]; b12.y = bp[Nc + 32];
        b13.x = bp[48]; b13.y = bp[Nc + 48];
        bp += bstep;
    }

    for (int it = 0; it < steps - 2; ++it) {
        // issue loads for step it+2 (oldest-in-flight stays 2 steps back)
        const v2f an = *reinterpret_cast<const v2f*>(ap);  ap += 4;
        v2f n0, n1, n2, n3;
        n0.x = bp[0];  n0.y = bp[Nc];
        n1.x = bp[16]; n1.y = bp[Nc + 16];
        n2.x = bp[32]; n2.y = bp[Nc + 32];
        n3.x = bp[48]; n3.y = bp[Nc + 48];
        bp += bstep;
        // consume step it (loads issued two iterations ago)
        acc0 = WMMA_F32(a0, b00, acc0);
        acc1 = WMMA_F32(a0, b01, acc1);
        acc2 = WMMA_F32(a0, b02, acc2);
        acc3 = WMMA_F32(a0, b03, acc3);
        // rotate (renamed away by unrolling)
        a0 = a1; b00 = b10; b01 = b11; b02 = b12; b03 = b13;
        a1 = an; b10 = n0;  b11 = n1;  b12 = n2;  b13 = n3;
    }
    // drain: step steps-2 (slot 0), then step steps-1 (slot 1)
    acc0 = WMMA_F32(a0, b00, acc0);
    acc1 = WMMA_F32(a0, b01, acc1);
    acc2 = WMMA_F32(a0, b02, acc2);
    acc3 = WMMA_F32(a0, b03, acc3);
    if (steps > 1) {
        acc0 = WMMA_F32(a1, b10, acc0);
        acc1 = WMMA_F32(a1, b11, acc1);
        acc2 = WMMA_F32(a1, b12, acc2);
        acc3 = WMMA_F32(a1, b13, acc3);
    }

    if (K4 != K) {                                // tail (K=74): clamped + select, no branches
        const int ka = K4 + kofs;
        const int kb = ka + 1;
        const float mka = (ka < K) ? 1.f : 0.f;
        const float mkb = (kb < K) ? 1.f : 0.f;
        const int cka = imin(ka, K - 1);
        const int ckb = imin(kb, K - 1);
        v2f at;
        at.x = A[(size_t)arow * K + cka] * mka;
        at.y = A[(size_t)arow * K + ckb] * mkb;
        const float* bra = Bm + (size_t)cka * Nc + col0 + l;
        const float* brb = Bm + (size_t)ckb * Nc + col0 + l;
        v2f t0, t1, t2, t3;
        t0.x = bra[0]  * mka; t0.y = brb[0]  * mkb;
        t1.x = bra[16] * mka; t1.y = brb[16] * mkb;
        t2.x = bra[32] * mka; t2.y = brb[32] * mkb;
        t3.x = bra[48] * mka; t3.y = brb[48] * mkb;
        acc0 = WMMA_F32(at, t0, acc0);
        acc1 = WMMA_F32(at, t1, acc1);
        acc2 = WMMA_F32(at, t2, acc2);
        acc3 = WMMA_F32(at, t3, acc3);
    }

    float bv0 = 0.f, bv1 = 0.f, bv2 = 0.f, bv3 = 0.f;
    if (bias) {
        bv0 = bias[col0 + l];
        bv1 = bias[col0 + 16 + l];
        bv2 = bias[col0 + 32 + l];
        bv3 = bias[col0 + 48 + l];
    }
    float* crow = C + (size_t)(row0 + (half << 3)) * Nc + col0 + l;
#pragma unroll
    for (int j = 0; j < 8; ++j) {                 // C layout: vgpr j -> row j (lanes 0-15) / j+8
        float v0 = acc0[j] + bv0;
        float v1 = acc1[j] + bv1;
        float v2 = acc2[j] + bv2;
        float v3 = acc3[j] + bv3;
        if (doRelu) {
            v0 = fmaxf(v0, 0.f); v1 = fmaxf(v1, 0.f);
            v2 = fmaxf(v2, 0.f); v3 = fmaxf(v3, 0.f);
        }
        crow[0] = v0; crow[16] = v1; crow[32] = v2; crow[48] = v3;
        crow += Nc;
    }
}

// =====================================================================
// GraphConv aggregation (norm='both'), one workgroup per molecule.
// aggOut[n] = rsqrt(d_in[n]) * sum_{e: dst=n} rsqrt(d_out[src]) * x[src]
// Edges are strictly intra-molecule -> everything lives in LDS, no
// global atomics; each thread owns one feature column (deterministic).
// =====================================================================
__global__ void gconv_agg(const float* __restrict__ x1, const float* __restrict__ x2,
                          const int* __restrict__ src1, const int* __restrict__ dst1,
                          const int* __restrict__ src2, const int* __restrict__ dst2,
                          float* __restrict__ aggOut, int F)
{
    __shared__ int   ls[EPM], ldv[EPM];
    __shared__ int   cOut[NPM], cIn[NPM];
    __shared__ float xs[NPM * HDIM];
    __shared__ float ag[NPM * HDIM];

    const int tid   = threadIdx.x;               // blockDim = 128
    const int blk   = blockIdx.x;                // [0, 2*BATCH)
    const int graph = (blk >= BATCH) ? 1 : 0;
    const int m     = blk - graph * BATCH;
    const float* x  = graph ? x2 : x1;
    const int* src  = graph ? src2 : src1;
    const int* dst  = graph ? dst2 : dst1;
    const int nbase = m * NPM;
    const int ebase = m * EPM;

    if (tid < EPM) {
        ls[tid]  = src[ebase + tid] - nbase;
        ldv[tid] = dst[ebase + tid] - nbase;
    }
    if (tid < NPM) { cOut[tid] = 0; cIn[tid] = 0; }
    __syncthreads();
    if (tid < EPM) {
        atomicAdd(&cOut[ls[tid]], 1);
        atomicAdd(&cIn[ldv[tid]], 1);
    }
    __syncthreads();

    for (int idx = tid; idx < NPM * F; idx += blockDim.x) {
        const int nl = idx / F;
        const int f  = idx - nl * F;
        xs[idx] = x[(size_t)(nbase + nl) * F + f] * rsqrtf((float)imax(cOut[nl], 1));
        ag[idx] = 0.f;
    }
    __syncthreads();

    if (tid < F) {                                // column-exclusive scatter-add
        for (int e = 0; e < EPM; ++e)
            ag[ldv[e] * F + tid] += xs[ls[e] * F + tid];
    }
    __syncthreads();

    const int gOff = graph * NNODE;
    for (int idx = tid; idx < NPM * F; idx += blockDim.x) {
        const int nl = idx / F;
        const int f  = idx - nl * F;
        aggOut[(size_t)(gOff + nbase + nl) * F + f] =
            ag[idx] * rsqrtf((float)imax(cIn[nl], 1));
    }
}

// mean over NPM nodes per molecule, times mixture fraction
__global__ void pool_kernel(const float* __restrict__ t,
                            const float* __restrict__ solv1_x,
                            float* __restrict__ xc)
{
    const int blk   = blockIdx.x;                 // [0, 2*BATCH)
    const int f     = threadIdx.x;                // 128
    const int graph = (blk >= BATCH) ? 1 : 0;
    const int i     = blk - graph * BATCH;
    const int base  = graph * NNODE + i * NPM;
    float s = 0.f;
    for (int r = 0; r < NPM; ++r) s += t[(size_t)(base + r) * HDIM + f];
    s *= (1.f / (float)NPM);
    const float sc = graph ? (1.f - solv1_x[i]) : solv1_x[i];
    xc[(size_t)blk * HDIM + f] = s * sc;
}

// We2[k, d*128+f]  ->  W2big[d, k*128+f]   (128 x 4096)
__global__ void repack_w2(const float* __restrict__ We2, float* __restrict__ W2big)
{
    const int idx = blockIdx.x * blockDim.x + threadIdx.x;
    if (idx >= HDIM * 32 * HDIM) return;
    const int d = idx >> 12;
    const int rem = idx & 4095;
    const int k = rem >> 7;
    const int f = rem & 127;
    W2big[idx] = We2[(size_t)k * (HDIM * HDIM) + d * HDIM + f];
}

// g = relu(hb * We1 + be1)  for the 3 edge-feature vectors -> (3, B, 32)
__global__ void edge_gate(const float* __restrict__ inter,
                          const float* __restrict__ ia1,
                          const float* __restrict__ ia2,
                          const float* __restrict__ We1,
                          const float* __restrict__ be1,
                          float* __restrict__ g3)
{
    const int idx = blockIdx.x * blockDim.x + threadIdx.x;
    if (idx >= 3 * BATCH * 32) return;
    const int which = idx / (BATCH * 32);
    const int rem   = idx - which * (BATCH * 32);
    const int i = rem >> 5;
    const int k = rem & 31;
    const float hb = (which == 0) ? inter[i] : (which == 1) ? ia1[i] : ia2[i];
    g3[idx] = fmaxf(hb * We1[k] + be1[k], 0.f);
}

// rbuf[n] = relu( bnn + Yb[i]+Yb[B+i] + sum_k gA[k]*Y[rA,k*128+f]
//                                     + sum_k gB[k]*Y[rB,k*128+f] )
__global__ void msg_kernel(const float* __restrict__ Y,
                           const float* __restrict__ Yb,
                           const float* __restrict__ g3,
                           const float* __restrict__ bnn,
                           float* __restrict__ rbuf)
{
    __shared__ float gA[32], gB[32];
    const int n = blockIdx.x;                     // [0, 2*BATCH)
    const int f = threadIdx.x;                    // 128
    const int lower = (n < BATCH);
    const int i = lower ? n : n - BATCH;
    const int rA = lower ? (BATCH + i) : i;                           // paired with g_inter
    const int rB = lower ? i : (BATCH + i);                           // paired with g_intra
    const float* ginter = g3 + (size_t)i * 32;
    const float* gintra = g3 + (size_t)(lower ? 1 : 2) * BATCH * 32 + (size_t)i * 32;
    if (f < 32) { gA[f] = ginter[f]; gB[f] = gintra[f]; }
    __syncthreads();
    float acc = bnn[f] + Yb[(size_t)rA * HDIM + f] + Yb[(size_t)rB * HDIM + f];
    const float* ya = Y + (size_t)rA * 4096 + f;
    const float* yb = Y + (size_t)rB * 4096 + f;
    for (int k = 0; k < 32; ++k)
        acc += gA[k] * ya[k * HDIM] + gB[k] * yb[k * HDIM];
    rbuf[(size_t)n * HDIM + f] = fmaxf(acc, 0.f);
}

// GRU cell elementwise part (gi, gh precomputed via GEMM)
__global__ void gru_kernel(const float* __restrict__ gi,
                           const float* __restrict__ gh,
                           const float* __restrict__ hstate,
                           float* __restrict__ xnew)
{
    const int idx = blockIdx.x * blockDim.x + threadIdx.x;
    if (idx >= 2 * BATCH * HDIM) return;
    const int row = idx >> 7;
    const int f   = idx & 127;
    const float* gir = gi + (size_t)row * 384;
    const float* ghr = gh + (size_t)row * 384;
    const float r = 1.f / (1.f + expf(-(gir[f] + ghr[f])));
    const float z = 1.f / (1.f + expf(-(gir[128 + f] + ghr[128 + f])));
    const float nn = tanhf(gir[256 + f] + r * ghr[256 + f]);
    xnew[idx] = (1.f - z) * nn + z * hstate[idx];
}

// hg[i] = concat(xnew[i], xnew[B+i])   (B x 256)
__global__ void concat_kernel(const float* __restrict__ xnew, float* __restrict__ hg)
{
    const int idx = blockIdx.x * blockDim.x + threadIdx.x;
    if (idx >= BATCH * 256) return;
    const int i = idx >> 8;
    const int c = idx & 255;
    hg[idx] = (c < HDIM) ? xnew[(size_t)i * HDIM + c]
                         : xnew[(size_t)(BATCH + i) * HDIM + (c - HDIM)];
}

// final tiny head: out[i,c] = o2[i,:] @ Wcl3[:,c] + bcl3[c]   (Nc=2)
__global__ void head_kernel(const float* __restrict__ o2,
                            const float* __restrict__ Wcl3,
                            const float* __restrict__ bcl3,
                            float* __restrict__ out)
{
    const int idx = blockIdx.x * blockDim.x + threadIdx.x;
    if (idx >= BATCH * NC) return;
    const int i = idx >> 1;
    const int c = idx & 1;
    float acc = bcl3[c];
    for (int j = 0; j < HDIM; ++j)
        acc += o2[(size_t)i * HDIM + j] * Wcl3[j * NC + c];
    out[idx] = acc;
}

// ---------------------------------------------------------------------
static inline void run_gemm(const float* A, const float* Bm, const float* bias,
                            float* C, int M, int Nc, int K, int relu,
                            hipStream_t s)
{
    const int tiles  = (M / 16) * (Nc / 64);      // 16x64 wave tiles
    const int blocks = (tiles + 7) / 8;           // 8 waves (wave32) per 256-thread block
    gemm_wmma_f32<<<blocks, 256, 0, s>>>(A, Bm, bias, C, M, Nc, K, relu);
}

extern "C" void kernel_launch(void* const* d_in, const int* in_sizes, int n_in,
                              void* d_out, int out_size, void* d_ws, size_t ws_size,
                              hipStream_t stream)
{
    (void)in_sizes; (void)n_in; (void)out_size; (void)ws_size;

    const float* h1       = (const float*)d_in[0];
    const float* h2       = (const float*)d_in[1];
    const float* solv1_x  = (const float*)d_in[2];
    const float* inter_hb = (const float*)d_in[3];
    const float* intra1   = (const float*)d_in[4];
    const float* intra2   = (const float*)d_in[5];
    const float* Wg1      = (const float*)d_in[6];
    const float* bg1      = (const float*)d_in[7];
    const float* Wg2      = (const float*)d_in[8];
    const float* bg2      = (const float*)d_in[9];
    const float* Wp       = (const float*)d_in[10];
    const float* bp       = (const float*)d_in[11];
    const float* We1      = (const float*)d_in[12];
    const float* be1      = (const float*)d_in[13];
    const float* We2      = (const float*)d_in[14];
    const float* be2      = (const float*)d_in[15];   // viewed as 128x128 row-major
    const float* bnn      = (const float*)d_in[16];
    const float* Wih      = (const float*)d_in[17];
    const float* bih      = (const float*)d_in[18];
    const float* Whh      = (const float*)d_in[19];
    const float* bhh      = (const float*)d_in[20];
    const float* Wcl1     = (const float*)d_in[21];
    const float* bcl1     = (const float*)d_in[22];
    const float* Wcl2     = (const float*)d_in[23];
    const float* bcl2     = (const float*)d_in[24];
    const float* Wcl3     = (const float*)d_in[25];
    const float* bcl3     = (const float*)d_in[26];
    const int*   src1     = (const int*)d_in[27];
    const int*   dst1     = (const int*)d_in[28];
    const int*   src2     = (const int*)d_in[29];
    const int*   dst2     = (const int*)d_in[30];
    float* out = (float*)d_out;

    // ---- workspace layout (floats) ----
    float* W = (float*)d_ws;
    size_t o = 0;
    float* bufA  = W + o; o += 16777216;      // aggregation buffer (2N x F), later Y (4096x4096)
    float* tbuf  = W + o; o += (size_t)2 * NNODE * HDIM;   // GraphConv outputs (2N x 128)
    float* xc    = W + o; o += 2 * BATCH * HDIM;
    float* xp    = W + o; o += 2 * BATCH * HDIM;
    float* W2big = W + o; o += HDIM * 32 * HDIM;
    float* Yb    = W + o; o += 2 * BATCH * HDIM;
    float* g3    = W + o; o += 3 * BATCH * 32;
    float* rbuf  = W + o; o += 2 * BATCH * HDIM;
    float* gi    = W + o; o += 2 * BATCH * 384;
    float* gh    = W + o; o += 2 * BATCH * 384;
    float* xnew  = W + o; o += 2 * BATCH * HDIM;
    float* hg    = W + o; o += BATCH * 256;
    float* o1    = W + o; o += BATCH * HDIM;
    float* o2    = W + o; o += BATCH * HDIM;

    // ---- GraphConv layer 1 (both graphs batched as 2N rows) ----
    gconv_agg<<<2 * BATCH, 128, 0, stream>>>(h1, h2, src1, dst1, src2, dst2, bufA, INDIM);
    run_gemm(bufA, Wg1, bg1, tbuf, 2 * NNODE, HDIM, INDIM, 1, stream);

    // ---- GraphConv layer 2 ----
    gconv_agg<<<2 * BATCH, 128, 0, stream>>>(tbuf, tbuf + (size_t)NNODE * HDIM,
                                             src1, dst1, src2, dst2, bufA, HDIM);
    run_gemm(bufA, Wg2, bg2, tbuf, 2 * NNODE, HDIM, HDIM, 1, stream);

    // ---- mean-pool + mixture fractions -> xc (4096 x 128) ----
    pool_kernel<<<2 * BATCH, 128, 0, stream>>>(tbuf, solv1_x, xc);

    // ---- projection: xp = relu(xc @ Wp + bp) ; xp is both x and hstate ----
    run_gemm(xc, Wp, bp, xp, 2 * BATCH, HDIM, HDIM, 1, stream);

    // ---- fused NNConv: Y = xp @ W2big (unique sources only), Yb = xp @ be2_mat ----
    repack_w2<<<(HDIM * 32 * HDIM + 255) / 256, 256, 0, stream>>>(We2, W2big);
    run_gemm(xp, W2big, nullptr, bufA, 2 * BATCH, 32 * HDIM, HDIM, 0, stream);  // Y in bufA
    run_gemm(xp, be2, nullptr, Yb, 2 * BATCH, HDIM, HDIM, 0, stream);

    // ---- edge gates + gated aggregation -> rbuf = relu(agg + bnn) ----
    edge_gate<<<(3 * BATCH * 32 + 255) / 256, 256, 0, stream>>>(inter_hb, intra1, intra2,
                                                                We1, be1, g3);
    msg_kernel<<<2 * BATCH, 128, 0, stream>>>(bufA, Yb, g3, bnn, rbuf);

    // ---- GRU step ----
    run_gemm(rbuf, Wih, bih, gi, 2 * BATCH, 3 * HDIM, HDIM, 0, stream);
    run_gemm(xp,   Whh, bhh, gh, 2 * BATCH, 3 * HDIM, HDIM, 0, stream);
    gru_kernel<<<(2 * BATCH * HDIM + 255) / 256, 256, 0, stream>>>(gi, gh, xp, xnew);

    // ---- classifier head ----
    concat_kernel<<<(BATCH * 256 + 255) / 256, 256, 0, stream>>>(xnew, hg);
    run_gemm(hg, Wcl1, bcl1, o1, BATCH, HDIM, 2 * HDIM, 1, stream);
    run_gemm(o1, Wcl2, bcl2, o2, BATCH, HDIM, HDIM, 1, stream);
    head_kernel<<<(BATCH * NC + 255) / 256, 256, 0, stream>>>(o2, Wcl3, bcl3, out);
}